// SpatialEncoder_790273983042
// MI455X (gfx1250) — compile-verified
//
#include <hip/hip_runtime.h>
#include <hip/hip_bf16.h>
#include <math.h>

// Problem constants (match reference)
#define NNODES 100000   // divisible by 16 -> no partial M tiles anywhere
#define EDGES  2000000
#define NGRAPH 256

typedef __attribute__((ext_vector_type(2))) float v2f;
typedef __attribute__((ext_vector_type(8))) float v8f;

// ---------------------------------------------------------------- utilities

__global__ void zero_f32_kernel(float* __restrict__ p, int n) {
    int i = blockIdx.x * blockDim.x + threadIdx.x;
    int stride = gridDim.x * blockDim.x;
    for (; i < n; i += stride) p[i] = 0.0f;
}

// deg[src[e]] += 1  (reference: segment_sum(ones, src))
__global__ void degree_kernel(const int* __restrict__ src, float* __restrict__ deg) {
    int e = blockIdx.x * blockDim.x + threadIdx.x;
    if (e < EDGES) atomicAdd(&deg[src[e]], 1.0f);
}

// dis = deg > 0 ? rsqrt(max(deg,1e-12)) : 0   (in place)
__global__ void rsqrt_kernel(float* __restrict__ d) {
    int i = blockIdx.x * blockDim.x + threadIdx.x;
    if (i < NNODES) {
        float v = d[i];
        d[i] = (v > 0.0f) ? rsqrtf(fmaxf(v, 1e-12f)) : 0.0f;
    }
}

// norm[e] = -dis[src]*dis[dst]  (precomputed once, reused by all 6 props)
__global__ void norm_kernel(const int* __restrict__ src, const int* __restrict__ dst,
                            const float* __restrict__ dis, float* __restrict__ norm) {
    int e = blockIdx.x * blockDim.x + threadIdx.x;
    if (e < EDGES) norm[e] = -dis[src[e]] * dis[dst[e]];
}

// x4[i*4 + f] = f<3 ? x[i*3+f] : 0   (pad layer-1 features to K-stride 4)
__global__ void pad_x_kernel(const float* __restrict__ x, float* __restrict__ x4) {
    int tid = blockIdx.x * blockDim.x + threadIdx.x;
    if (tid < NNODES * 4) {
        int i = tid >> 2, f = tid & 3;
        x4[tid] = (f < 3) ? x[i * 3 + f] : 0.0f;
    }
}

// W1p (3,4,16) = zero-padded W1 (3,3,16)
__global__ void pad_w1_kernel(const float* __restrict__ w1, float* __restrict__ w1p) {
    int tid = blockIdx.x * blockDim.x + threadIdx.x;
    if (tid < 3 * 4 * 16) {
        int t = tid >> 6;          // hop
        int r = tid & 63;
        int k = r >> 4, n = r & 15;
        w1p[tid] = (k < 3) ? w1[t * 48 + k * 16 + n] : 0.0f;
    }
}

// ---------------------------------------------------------------- propagation
// hout[dst*S + f] += norm[e] * hin[src*S + f]  for f < F; TPE threads per edge.
template <int F, int S, int TPE>
__global__ void prop_kernel(const int* __restrict__ src, const int* __restrict__ dst,
                            const float* __restrict__ norm,
                            const float* __restrict__ hin, float* __restrict__ hout) {
    int tid = blockIdx.x * blockDim.x + threadIdx.x;
    int e = tid / TPE;
    if (e >= EDGES) return;
    int part = tid - e * TPE;
    int s = src[e], d = dst[e];
    float w = norm[e];
    constexpr int FP = F / TPE;                 // features per thread
    int bs = s * S + part * FP;
    int bd = d * S + part * FP;
    if constexpr (FP == 4) {                    // F=16/32: 16B-aligned chunks
        float4 v = *reinterpret_cast<const float4*>(hin + bs);
        atomicAdd(&hout[bd + 0], w * v.x);
        atomicAdd(&hout[bd + 1], w * v.y);
        atomicAdd(&hout[bd + 2], w * v.z);
        atomicAdd(&hout[bd + 3], w * v.w);
    } else if constexpr (FP == 3 && S == 4) {   // layer 1: padded stride-4 buffer
        float4 v = *reinterpret_cast<const float4*>(hin + bs);
        atomicAdd(&hout[bd + 0], w * v.x);
        atomicAdd(&hout[bd + 1], w * v.y);
        atomicAdd(&hout[bd + 2], w * v.z);
    } else {
#pragma unroll
        for (int f = 0; f < FP; ++f)
            atomicAdd(&hout[bd + f], w * hin[bs + f]);
    }
}

// Tx2 = 2*P - Tx0   (elementwise, after P = prop(Tx1))
__global__ void cheb2_kernel(float* __restrict__ tx2, const float* __restrict__ tx0, int n) {
    int i = blockIdx.x * blockDim.x + threadIdx.x;
    if (i < n) tx2[i] = 2.0f * tx2[i] - tx0[i];
}

// ---------------------------------------------------------------- dense combine via WMMA
// out[N x FOUT] = act( Tx0@W[0] + Tx1@W[1] + Tx2@W[2] + b ), Tx* stride KS (mult of 4),
// W shape (3, KS, FOUT) (zero-padded rows contribute nothing).
// One wave32 per 16x16 output tile; K accumulated with V_WMMA_F32_16X16X4_F32.
// A (16x4 f32): lanes 0-15 & 16-31 both hold rows M=0..15; lane half selects K pair
//   {0,1} vs {2,3} across the two A VGPRs -> one aligned v2f load per step, no masking.
// B (4x16 f32): lane holds column, same K split. C/D: VGPR r -> row r (+8 for hi half).
template <int KS, int FOUT, int ACT>   // ACT: 0 = PReLU(0.2), 1 = tanh
__global__ __launch_bounds__(32) void cheb_wmma_kernel(
    const float* __restrict__ tx0, const float* __restrict__ tx1,
    const float* __restrict__ tx2, const float* __restrict__ W,
    const float* __restrict__ bias, float* __restrict__ out) {
    constexpr int NT = FOUT / 16;
    const int tile = blockIdx.x;
    const int mt = tile / NT;
    const int nt = tile - mt * NT;
    const int m0 = mt * 16, n0 = nt * 16;
    const int lane = threadIdx.x;     // 0..31, full wave, EXEC all-1s throughout
    const int half = lane >> 4;       // 0: K {0,1}; 1: K {2,3} within each 4-wide block
    const int l = lane & 15;
    const int arow = m0 + l;          // always < NNODES (N % 16 == 0)
    const int col = n0 + l;

    v8f acc = {};
#pragma unroll
    for (int t = 0; t < 3; ++t) {
        const float* __restrict__ T  = (t == 0) ? tx0 : (t == 1) ? tx1 : tx2;
        const float* __restrict__ Wt = W + t * KS * FOUT;
#pragma unroll
        for (int kb = 0; kb < KS / 4; ++kb) {
            const int k0 = kb * 4 + half * 2;
            v2f a = *reinterpret_cast<const v2f*>(T + arow * KS + k0);  // 8B aligned
            v2f b;
            b.x = Wt[k0 * FOUT + col];
            b.y = Wt[(k0 + 1) * FOUT + col];
            acc = __builtin_amdgcn_wmma_f32_16x16x4_f32(
                false, a, false, b, (short)0, acc, false, false);
        }
    }

    const float bc = bias[col];
#pragma unroll
    for (int r = 0; r < 8; ++r) {
        const int orow = m0 + half * 8 + r;   // always < NNODES
        float v = acc[r] + bc;
        if (ACT == 0) v = (v >= 0.0f) ? v : 0.2f * v;
        else          v = tanhf(v);
        out[orow * FOUT + col] = v;
    }
}

// ---------------------------------------------------------------- global add pool
__global__ void pool_kernel(const float* __restrict__ h, const int* __restrict__ batch,
                            float* __restrict__ out) {
    int tid = blockIdx.x * blockDim.x + threadIdx.x;
    if (tid >= NNODES * 64) return;
    int i = tid >> 6;
    int f = tid & 63;
    atomicAdd(&out[batch[i] * 64 + f], h[tid]);
}

// ---------------------------------------------------------------- launch

static inline int cdiv(long a, int b) { return (int)((a + b - 1) / b); }

extern "C" void kernel_launch(void* const* d_in, const int* in_sizes, int n_in,
                              void* d_out, int out_size, void* d_ws, size_t ws_size,
                              hipStream_t stream) {
    const float* x     = (const float*)d_in[0];        // [N,3]
    const int*   ei    = (const int*)d_in[1];          // [2,E]
    const int*   batch = (const int*)d_in[2];          // [N]
    const float* W1    = (const float*)d_in[3];        // [3,3,16]
    const float* b1    = (const float*)d_in[4];
    const float* W2    = (const float*)d_in[5];        // [3,16,32]
    const float* b2    = (const float*)d_in[6];
    const float* W3    = (const float*)d_in[7];        // [3,32,64]
    const float* b3    = (const float*)d_in[8];
    float* out = (float*)d_out;                        // [256,64]

    const int* src = ei;
    const int* dst = ei + EDGES;

    // ws layout (floats):
    // dis[N] | norm[E] | HA[N*64] | HB[N*64] | Tx1[N*32] | Tx2[N*32] | X4[N*4] | W1p[192]
    float* ws   = (float*)d_ws;
    float* dis  = ws;
    float* nrm  = dis + NNODES;
    float* HA   = nrm + EDGES;
    float* HB   = HA + (size_t)NNODES * 64;
    float* Tx1  = HB + (size_t)NNODES * 64;
    float* Tx2  = Tx1 + (size_t)NNODES * 32;
    float* X4   = Tx2 + (size_t)NNODES * 32;
    float* W1p  = X4 + (size_t)NNODES * 4;

    const int ZB = 256;
    auto zero = [&](float* p, long n) {
        int blocks = cdiv(n, ZB); if (blocks > 16384) blocks = 16384;
        zero_f32_kernel<<<blocks, ZB, 0, stream>>>(p, (int)n);
    };

    // --- normalization: deg -> dis -> per-edge norm
    zero(dis, NNODES);
    degree_kernel<<<cdiv(EDGES, ZB), ZB, 0, stream>>>(src, dis);
    rsqrt_kernel<<<cdiv(NNODES, ZB), ZB, 0, stream>>>(dis);
    norm_kernel<<<cdiv(EDGES, ZB), ZB, 0, stream>>>(src, dst, dis, nrm);

    // --- pad layer-1 operands to K-stride 4
    pad_x_kernel<<<cdiv((long)NNODES * 4, ZB), ZB, 0, stream>>>(x, X4);
    pad_w1_kernel<<<1, 256, 0, stream>>>(W1, W1p);

    // --- layer 1: K-stride 4 (3 real feats), Fout=16, PReLU -> HA
    zero(Tx1, (long)NNODES * 4);
    prop_kernel<3, 4, 1><<<cdiv((long)EDGES, ZB), ZB, 0, stream>>>(src, dst, nrm, X4, Tx1);
    zero(Tx2, (long)NNODES * 4);
    prop_kernel<3, 4, 1><<<cdiv((long)EDGES, ZB), ZB, 0, stream>>>(src, dst, nrm, Tx1, Tx2);
    cheb2_kernel<<<cdiv((long)NNODES * 4, ZB), ZB, 0, stream>>>(Tx2, X4, NNODES * 4);
    cheb_wmma_kernel<4, 16, 0><<<(NNODES / 16) * 1, 32, 0, stream>>>(X4, Tx1, Tx2, W1p, b1, HA);

    // --- layer 2: Fin=16 (Tx0 = HA), Fout=32, PReLU -> HB
    zero(Tx1, (long)NNODES * 16);
    prop_kernel<16, 16, 4><<<cdiv((long)EDGES * 4, ZB), ZB, 0, stream>>>(src, dst, nrm, HA, Tx1);
    zero(Tx2, (long)NNODES * 16);
    prop_kernel<16, 16, 4><<<cdiv((long)EDGES * 4, ZB), ZB, 0, stream>>>(src, dst, nrm, Tx1, Tx2);
    cheb2_kernel<<<cdiv((long)NNODES * 16, ZB), ZB, 0, stream>>>(Tx2, HA, NNODES * 16);
    cheb_wmma_kernel<16, 32, 0><<<(NNODES / 16) * 2, 32, 0, stream>>>(HA, Tx1, Tx2, W2, b2, HB);

    // --- layer 3: Fin=32 (Tx0 = HB), Fout=64, tanh -> HA
    zero(Tx1, (long)NNODES * 32);
    prop_kernel<32, 32, 8><<<cdiv((long)EDGES * 8, ZB), ZB, 0, stream>>>(src, dst, nrm, HB, Tx1);
    zero(Tx2, (long)NNODES * 32);
    prop_kernel<32, 32, 8><<<cdiv((long)EDGES * 8, ZB), ZB, 0, stream>>>(src, dst, nrm, Tx1, Tx2);
    cheb2_kernel<<<cdiv((long)NNODES * 32, ZB), ZB, 0, stream>>>(Tx2, HB, NNODES * 32);
    cheb_wmma_kernel<32, 64, 1><<<(NNODES / 16) * 4, 32, 0, stream>>>(HB, Tx1, Tx2, W3, b3, HA);

    // --- global add pool -> out[256,64]
    zero(out, (long)NGRAPH * 64);
    pool_kernel<<<cdiv((long)NNODES * 64, ZB), ZB, 0, stream>>>(HA, batch, out);
}